// Attn_13846974562399
// MI455X (gfx1250) — compile-verified
//
#include <hip/hip_runtime.h>
#include <hip/hip_bf16.h>

#define SS 2048
#define BB 32
#define HH 1024

typedef float v2f __attribute__((ext_vector_type(2)));
typedef float v8f __attribute__((ext_vector_type(8)));

// -----------------------------------------------------------------------------
// Stage 1: u[b,h] = sum_k hidden[b,k] * W[k,h]   (32 x 1024 x 1024 fp32 GEMM)
// One wave (32 lanes) per 16x16 output tile; K-loop of V_WMMA_F32_16X16X4_F32.
// A(16x4) lane map: lanes 0-15 -> M=lane, K = kb+{0,1}; lanes 16-31 -> K = kb+{2,3}.
// B(4x16) lane map: lanes 0-15 -> N=lane, K = kb+{0,1}; lanes 16-31 -> N=lane-16, K = kb+{2,3}.
// D lane map: VGPR r -> M = r (lanes 0-15) / r+8 (lanes 16-31), N = lane&15.
// -----------------------------------------------------------------------------
__global__ __launch_bounds__(32) void attn_uproj_wmma(
    const float* __restrict__ hidden,  // [BB, HH]
    const float* __restrict__ W,       // [HH, HH]  (W[k,h])
    float* __restrict__ u)             // [BB, HH]
{
    const int lane = threadIdx.x & 31;
    const int n0   = blockIdx.x * 16;          // h-tile
    const int m0   = blockIdx.y * 16;          // b-tile
    const int mr   = lane & 15;
    const int kOff = (lane >> 4) << 1;         // 0 for lanes 0-15, 2 for lanes 16-31

    const float* Arow = hidden + (size_t)(m0 + mr) * HH;  // hidden[m, :]
    const float* Bcol = W + (n0 + mr);                    // W[:, n] (stride HH)

    v8f c = {0.f, 0.f, 0.f, 0.f, 0.f, 0.f, 0.f, 0.f};
    for (int kb = 0; kb < HH; kb += 4) {
        v2f a, b;
        a.x = Arow[kb + kOff];
        a.y = Arow[kb + kOff + 1];
        b.x = Bcol[(size_t)(kb + kOff) * HH];
        b.y = Bcol[(size_t)(kb + kOff + 1) * HH];
        c = __builtin_amdgcn_wmma_f32_16x16x4_f32(
                /*neg_a=*/false, a, /*neg_b=*/false, b,
                /*c_mod=*/(short)0, c, /*reuse_a=*/false, /*reuse_b=*/false);
    }

    const int nOut  = n0 + mr;
    const int mBase = m0 + (lane >> 4) * 8;
    #pragma unroll
    for (int r = 0; r < 8; ++r)
        u[(size_t)(mBase + r) * HH + nOut] = c[r];
}

// -----------------------------------------------------------------------------
// Stage 2: scores[b,s] = u[b] . enc[s,b,:]   (HBM-bound: streams 268 MB once)
// Block = 256 threads (8 waves), one (b, 128-wide s-chunk) per block.
// Each lane caches its 32 fixed u-values in VGPRs (no LDS in the hot loop),
// then streams enc rows with float4 (global_load_b128) loads: 512 B contiguous
// per wave-iteration, pure load+v_fmac inner loop, wave32 shuffle reduction.
// Scores written straight to d_out.
// -----------------------------------------------------------------------------
__global__ __launch_bounds__(256) void attn_scores(
    const float* __restrict__ enc,     // [SS, BB, HH]
    const float* __restrict__ u,       // [BB, HH]
    float* __restrict__ scores)        // [BB, SS] (== d_out)
{
    const int b    = blockIdx.y;
    const int tid  = threadIdx.x;
    const int wave = tid >> 5;
    const int lane = tid & 31;

    // Hoist this lane's slice of u into registers: u4[lane + j*32], j = 0..7
    const float4* u4 = (const float4*)(u + (size_t)b * HH);
    float4 uR[8];
    #pragma unroll
    for (int j = 0; j < 8; ++j)
        uR[j] = u4[lane + j * 32];

    const int sBase = blockIdx.x * 128;

    for (int i = 0; i < 16; ++i) {
        const int s = sBase + i * 8 + wave;
        const float4* p = (const float4*)(enc + ((size_t)s * BB + b) * HH);

        if (i < 15) {
            // prefetch next row for this wave (gfx1250 global_prefetch_b8)
            const float* pn = enc + ((size_t)(s + 8) * BB + b) * HH;
            __builtin_prefetch((const void*)(pn + lane * 32), 0, 0);
        }

        float acc = 0.f;
        #pragma unroll
        for (int j = 0; j < 8; ++j) {
            float4 e = p[lane + j * 32];
            acc = fmaf(e.x, uR[j].x, acc);
            acc = fmaf(e.y, uR[j].y, acc);
            acc = fmaf(e.z, uR[j].z, acc);
            acc = fmaf(e.w, uR[j].w, acc);
        }
        #pragma unroll
        for (int off = 16; off > 0; off >>= 1)
            acc += __shfl_xor(acc, off, 32);
        if (lane == 0)
            scores[(size_t)b * SS + s] = acc;
    }
}

// -----------------------------------------------------------------------------
// Stage 3: in-place softmax over s for each b. (hidden·bias constant term is
// softmax-shift-invariant, so it is omitted; bias is zero in setup anyway.)
// -----------------------------------------------------------------------------
__global__ __launch_bounds__(256) void attn_softmax(float* __restrict__ out)
{
    __shared__ float red[8];
    const int b    = blockIdx.x;
    float* row     = out + (size_t)b * SS;
    const int tid  = threadIdx.x;
    const int lane = tid & 31;
    const int wave = tid >> 5;

    // --- max ---
    float m = -3.402823466e+38f;
    for (int s = tid; s < SS; s += 256) m = fmaxf(m, row[s]);
    #pragma unroll
    for (int off = 16; off > 0; off >>= 1) m = fmaxf(m, __shfl_xor(m, off, 32));
    if (lane == 0) red[wave] = m;
    __syncthreads();
    float M = fmaxf(fmaxf(fmaxf(red[0], red[1]), fmaxf(red[2], red[3])),
                    fmaxf(fmaxf(red[4], red[5]), fmaxf(red[6], red[7])));
    __syncthreads();

    // --- exp + sum ---
    float sum = 0.f;
    for (int s = tid; s < SS; s += 256) {
        float e = __expf(row[s] - M);
        row[s] = e;
        sum += e;
    }
    #pragma unroll
    for (int off = 16; off > 0; off >>= 1) sum += __shfl_xor(sum, off, 32);
    if (lane == 0) red[wave] = sum;
    __syncthreads();
    float SUM = (red[0] + red[1]) + (red[2] + red[3]) +
                (red[4] + red[5]) + (red[6] + red[7]);
    const float inv = 1.0f / SUM;

    // --- normalize ---
    for (int s = tid; s < SS; s += 256) row[s] *= inv;
}

extern "C" void kernel_launch(void* const* d_in, const int* in_sizes, int n_in,
                              void* d_out, int out_size, void* d_ws, size_t ws_size,
                              hipStream_t stream) {
    const float* hidden = (const float*)d_in[0];  // [1, BB, HH]
    const float* enc    = (const float*)d_in[1];  // [SS, BB, HH]
    const float* W      = (const float*)d_in[2];  // [HH, HH]
    // d_in[3] = bias: contributes a per-b constant to scores -> softmax-invariant.

    float* u   = (float*)d_ws;    // BB*HH floats = 128 KB scratch
    float* out = (float*)d_out;   // BB*SS floats: scores, then attn in place

    attn_uproj_wmma<<<dim3(HH / 16, BB / 16), 32, 0, stream>>>(hidden, W, u);
    attn_scores  <<<dim3(SS / 128, BB), 256, 0, stream>>>(enc, u, out);
    attn_softmax <<<BB, 256, 0, stream>>>(out);
}